// MinibatchDiscrimination_80152679678415
// MI455X (gfx1250) — compile-verified
//
#include <hip/hip_runtime.h>
#include <hip/hip_bf16.h>

// MinibatchDiscrimination for MI455X (gfx1250, wave32).
// Phase 0: pad+transpose T [512,500] -> Wt [512 cols][512 k] (zeros in pad).
// Phase 1: M = x @ T via V_WMMA_F32_16X16X4_F32 (full f32 tensor path),
//          branch-free inner loop (2x b64 loads + wmma).
// Phase 2: out[j,o] = sum_i exp(-L1(M[i,o,:], M[j,o,:])) - 1, with i-rows
//          staged through LDS via GLOBAL_LOAD_ASYNC_TO_LDS_B128 (ASYNCcnt).

typedef __attribute__((ext_vector_type(2))) float v2f;
typedef __attribute__((ext_vector_type(8))) float v8f;

#define BATCH   512
#define IN_F    512
#define OUT_F   100
#define KDIM    5
#define NCOL    (OUT_F * KDIM)   // 500 real columns
#define MPAD    512              // padded stride

// ---------------------------------------------------------------------------
// Phase 0: Wt[c][k] = (c < 500) ? T[k][c] : 0    (K-major for b64 B loads)
// ---------------------------------------------------------------------------
__global__ __launch_bounds__(256)
void mbd_pad_w(const float* __restrict__ T, float* __restrict__ Wt) {
    const int k = blockIdx.x;                 // 0..511 (K row of T)
    for (int c = threadIdx.x; c < MPAD; c += 256)
        Wt[c * MPAD + k] = (c < NCOL) ? T[k * NCOL + c] : 0.0f;
}

// ---------------------------------------------------------------------------
// Phase 1: one wave per 16x16 C tile, grid 32x32.
// f32 WMMA fragment layouts per CDNA5 ISA 7.12.2:
//   A 16x4 : lane<16 -> M=lane, K={0,1}; lane>=16 -> M=lane-16, K={2,3}
//   B 4x16 : N = lane&15, row K = vgpr + 2*(lane>=16)
//   C 16x16: N = lane&15, row M = vgpr + 8*(lane>=16)
// Both A and B fragments are contiguous K-pairs -> single b64 load each.
// ---------------------------------------------------------------------------
__global__ __launch_bounds__(32)
void mbd_gemm_wmma(const float* __restrict__ x,
                   const float* __restrict__ Wt,
                   float* __restrict__ M) {
    const int lane = threadIdx.x;      // full wave, EXEC all-ones
    const int hi   = lane >> 4;
    const int lo   = lane & 15;
    const int row  = blockIdx.x * 16 + lo;   // A row for this lane
    const int col  = blockIdx.y * 16 + lo;   // B/C column for this lane

    const float* __restrict__ arow = x  + row * IN_F;
    const float* __restrict__ bcol = Wt + col * MPAD;

    v8f c = {};
    for (int k0 = 0; k0 < IN_F; k0 += 4) {
        const int ka = k0 + 2 * hi;          // this lane's K pair
        const float2 av = *(const float2*)(arow + ka);
        const float2 bv = *(const float2*)(bcol + ka);
        v2f a; a.x = av.x; a.y = av.y;
        v2f b; b.x = bv.x; b.y = bv.y;
        c = __builtin_amdgcn_wmma_f32_16x16x4_f32(
                false, a, false, b, (short)0, c, false, false);
    }

#pragma unroll
    for (int r = 0; r < 8; ++r)
        M[(blockIdx.x * 16 + r + 8 * hi) * MPAD + col] = c[r];
}

// ---------------------------------------------------------------------------
// Phase 2: one block per j, 128 threads; thread t handles feature o=t (o<100).
// i-rows staged 16 at a time (32 KB) into LDS with async copies: each of the
// 128 threads issues exactly 16 global_load_async_to_lds_b128 (2048 float4),
// then s_wait_asynccnt 0 + barrier. LDS columns o*5+k hit distinct banks
// across a wave (gcd(5,64)=1) so the compute loop is conflict-free.
// ---------------------------------------------------------------------------
#define CH 16
__global__ __launch_bounds__(128)
void mbd_pairwise(const float* __restrict__ M, float* __restrict__ out) {
    __shared__ __align__(16) float sM[CH * MPAD];    // 32 KB
    const int j = blockIdx.x;
    const int t = threadIdx.x;
    const int o = t;

    // LDS byte offset of sM: low 32 bits of the generic pointer (LDS aperture
    // keeps the offset in [31:0] on CDNA5).
    const unsigned sBase = (unsigned)(unsigned long long)(uintptr_t)sM;

    float mj[KDIM];
    if (o < OUT_F) {
#pragma unroll
        for (int k = 0; k < KDIM; ++k) mj[k] = M[j * MPAD + o * KDIM + k];
    }

    float acc = 0.0f;
    for (int i0 = 0; i0 < BATCH; i0 += CH) {
        const unsigned long long gBase =
            (unsigned long long)(uintptr_t)(M + i0 * MPAD);
#pragma unroll
        for (int s = 0; s < (CH * MPAD / 4) / 128; ++s) {   // 16 iterations
            const int idx = t + s * 128;                    // float4 index
            const unsigned           ldsAddr = sBase + (unsigned)idx * 16u;
            const unsigned long long gAddr   = gBase + (unsigned long long)idx * 16u;
            asm volatile("global_load_async_to_lds_b128 %0, %1, off"
                         :: "v"(ldsAddr), "v"(gAddr)
                         : "memory");
        }
        asm volatile("s_wait_asynccnt 0x0" ::: "memory");
        __syncthreads();

        if (o < OUT_F) {
#pragma unroll 4
            for (int r = 0; r < CH; ++r) {
                float n = 0.0f;
#pragma unroll
                for (int k = 0; k < KDIM; ++k)
                    n += fabsf(sM[r * MPAD + o * KDIM + k] - mj[k]);
                acc += __expf(-n);
            }
        }
        __syncthreads();   // protect sM before next chunk's asyncs land
    }

    if (o < OUT_F) out[j * OUT_F + o] = acc - 1.0f;  // -1 removes self term
}

// ---------------------------------------------------------------------------
extern "C" void kernel_launch(void* const* d_in, const int* in_sizes, int n_in,
                              void* d_out, int out_size, void* d_ws, size_t ws_size,
                              hipStream_t stream) {
    const float* x = (const float*)d_in[0];   // [512, 512]
    const float* T = (const float*)d_in[1];   // [512, 100, 5] = [512, 500]
    float* Wt  = (float*)d_ws;                          // [512, 512], 1 MB
    float* M   = Wt + MPAD * MPAD;                      // [512, 512], 1 MB
    float* out = (float*)d_out;                         // [512, 100]

    mbd_pad_w<<<IN_F, 256, 0, stream>>>(T, Wt);
    dim3 g1(BATCH / 16, MPAD / 16);                     // 32 x 32 tiles
    mbd_gemm_wmma<<<g1, 32, 0, stream>>>(x, Wt, M);
    mbd_pairwise<<<BATCH, 128, 0, stream>>>(M, out);
}